// DiagonalSelectiveCell_9981503995955
// MI455X (gfx1250) — compile-verified
//
#include <hip/hip_runtime.h>

#define T_STEPS 2048
#define BATCH   16
#define DIM     1024
#define NGROUPS 32
#define GSIZE   32                 // DIM / NGROUPS
#define MROWS   (T_STEPS * BATCH)  // 32768
#define BD      (BATCH * DIM)

typedef float v2f __attribute__((ext_vector_type(2)));
typedef float v8f __attribute__((ext_vector_type(8)));

static __device__ __forceinline__ v8f wmma4_f32(v2f a, v2f b, v8f c) {
  // V_WMMA_F32_16X16X4_F32 : D = A(16x4) * B(4x16) + C, full fp32
  return __builtin_amdgcn_wmma_f32_16x16x4_f32(false, a, false, b,
                                               (short)0, c, false, false);
}

// ---------------------------------------------------------------------------
// GEMM: C[m,n] = sum_k A[m,k] * W[n,k]   (A: MROWS x DIM, W: DIM x DIM)
// mode 0: dst = acc + bias[n]                         (cand_in)
// mode 1: dst = sigmoid(acc + bias[n])                (delta)
// mode 2: dst = exp(h-mx)/se * silu(acc)              (final output)
// Block = 256 thr = 8 waves (4 along M, 2 along N); block tile 128x128,
// wave tile 32(M) x 64(N) -> acc[2][4] 16x16 WMMA accumulators.
// ---------------------------------------------------------------------------
__global__ __launch_bounds__(256)
void dsc_gemm_kernel(const float* __restrict__ A, const float* __restrict__ W,
                     const float* __restrict__ bias,
                     const float* __restrict__ hsrc,
                     const float2* __restrict__ stats,
                     float* __restrict__ dst, int mode)
{
  const int lane = threadIdx.x & 31;
  const int wave = threadIdx.x >> 5;
  const int wm   = wave & 3;   // wave's M slot (0..3)
  const int wn   = wave >> 2;  // wave's N slot (0..1)
  const int lh   = lane & 15;  // row/col within 16
  const int kh   = lane >> 4;  // K pair select (0 or 1)

  const int m0 = blockIdx.x * 128 + wm * 32;
  const int n0 = blockIdx.y * 128 + wn * 64;

  // f32 WMMA A-frag layout: lane(l&15)=M row, VGPR pair holds K = kh*2, kh*2+1
  const float* arow0 = A + (m0 +      lh) * DIM + kh * 2;
  const float* arow1 = A + (m0 + 16 + lh) * DIM + kh * 2;
  // B-frag (4x16 = K x N): lane(l&15)=N col, K = kh*2 + {0,1}; B[k,n] = W[n,k]
  const float* brow0 = W + (n0 +      lh) * DIM + kh * 2;
  const float* brow1 = W + (n0 + 16 + lh) * DIM + kh * 2;
  const float* brow2 = W + (n0 + 32 + lh) * DIM + kh * 2;
  const float* brow3 = W + (n0 + 48 + lh) * DIM + kh * 2;

  v8f acc[2][4];
  const v8f vzero = {0.f, 0.f, 0.f, 0.f, 0.f, 0.f, 0.f, 0.f};
#pragma unroll
  for (int mi = 0; mi < 2; ++mi)
#pragma unroll
    for (int ni = 0; ni < 4; ++ni) acc[mi][ni] = vzero;

  for (int k0 = 0; k0 < DIM; k0 += 64) {
    // A is streamed from HBM; prefetch the next K-panel (gfx1250 global_prefetch_b8)
    if (k0 + 256 < DIM) {
      __builtin_prefetch(arow0 + k0 + 256, 0, 1);
      __builtin_prefetch(arow1 + k0 + 256, 0, 1);
    }
#pragma unroll 4
    for (int k = k0; k < k0 + 64; k += 4) {
      v2f a0 = *(const v2f*)(arow0 + k);
      v2f a1 = *(const v2f*)(arow1 + k);
      v2f b0 = *(const v2f*)(brow0 + k);
      v2f b1 = *(const v2f*)(brow1 + k);
      v2f b2 = *(const v2f*)(brow2 + k);
      v2f b3 = *(const v2f*)(brow3 + k);
      acc[0][0] = wmma4_f32(a0, b0, acc[0][0]);
      acc[0][1] = wmma4_f32(a0, b1, acc[0][1]);
      acc[0][2] = wmma4_f32(a0, b2, acc[0][2]);
      acc[0][3] = wmma4_f32(a0, b3, acc[0][3]);
      acc[1][0] = wmma4_f32(a1, b0, acc[1][0]);
      acc[1][1] = wmma4_f32(a1, b1, acc[1][1]);
      acc[1][2] = wmma4_f32(a1, b2, acc[1][2]);
      acc[1][3] = wmma4_f32(a1, b3, acc[1][3]);
    }
  }

  // C/D layout: VGPR j -> M = j + (lane>=16 ? 8 : 0), N = lane & 15
#pragma unroll
  for (int mi = 0; mi < 2; ++mi) {
    const int rbase = m0 + mi * 16 + (kh ? 8 : 0);
#pragma unroll
    for (int ni = 0; ni < 4; ++ni) {
      const int col = n0 + ni * 16 + lh;
      const float bv = (mode == 2) ? 0.0f : bias[col];
#pragma unroll
      for (int j = 0; j < 8; ++j) {
        const int row = rbase + j;
        float v = acc[mi][ni][j];
        if (mode == 0) {
          v += bv;
        } else if (mode == 1) {
          v += bv;
          v = 1.0f / (1.0f + expf(-v));
        } else {
          const float  hv = hsrc[row * DIM + col];
          const float2 st = stats[row * NGROUPS + (col >> 5)];
          const float compete = expf(hv - st.x) / st.y;
          v = compete * (v / (1.0f + expf(-v)));   // compete * silu(acc)
        }
        dst[row * DIM + col] = v;
      }
    }
  }
}

// ---------------------------------------------------------------------------
// Sequential scan: one wave32 == one (batch, group); lane == channel in group.
// delta lives in-place in the h_all[1:] region (read delta[t], overwrite with
// h[t+1] at the same address).  Emits per-group softmax stats (max, sum-exp).
// ---------------------------------------------------------------------------
__global__ __launch_bounds__(256)
void dsc_scan_kernel(const float* __restrict__ h0, const float* __restrict__ r_h,
                     float* __restrict__ h_all, const float* __restrict__ cand,
                     float2* __restrict__ stats)
{
  const int wid  = (int)((blockIdx.x * blockDim.x + threadIdx.x) >> 5);
  const int lane = threadIdx.x & 31;
  const int b    = wid >> 5;           // wid / NGROUPS
  const int g    = wid & 31;           // wid % NGROUPS
  const int d    = g * GSIZE + lane;
  const int base = b * DIM + d;

  const float rh = r_h[d];
  float h = h0[base];
  h_all[base] = h;                     // h_all[0] = h0

  // software pipeline: keep next step's (delta, cand) in flight
  float dlt = h_all[BD + base];        // delta[0] (stored at h slot t=1)
  float c   = cand[base];

  for (int t = 0; t < T_STEPS; ++t) {
    float dlt_n = 0.0f, c_n = 0.0f;
    if (t + 1 < T_STEPS) {
      dlt_n = h_all[(t + 2) * BD + base];
      c_n   = cand[(t + 1) * BD + base];
    }
    const float cd = tanhf(fmaf(rh, h, c));
    h = fmaf(dlt, cd - h, h);          // (1-d)*h + d*cand
    h_all[(t + 1) * BD + base] = h;    // overwrites consumed delta[t]

    // group softmax stats across the 32 lanes of this wave
    float mx = h;
#pragma unroll
    for (int off = 16; off > 0; off >>= 1)
      mx = fmaxf(mx, __shfl_xor(mx, off, 32));
    float se = expf(h - mx);
#pragma unroll
    for (int off = 16; off > 0; off >>= 1)
      se += __shfl_xor(se, off, 32);
    if (lane == 0)
      stats[(t * BATCH + b) * NGROUPS + g] = make_float2(mx, se);

    dlt = dlt_n;
    c   = c_n;
  }
}

// ---------------------------------------------------------------------------
extern "C" void kernel_launch(void* const* d_in, const int* in_sizes, int n_in,
                              void* d_out, int out_size, void* d_ws, size_t ws_size,
                              hipStream_t stream)
{
  const float* x       = (const float*)d_in[0];
  const float* h0      = (const float*)d_in[1];
  const float* W_x     = (const float*)d_in[2];
  const float* r_h     = (const float*)d_in[3];
  const float* b       = (const float*)d_in[4];
  const float* W_delta = (const float*)d_in[5];
  const float* b_delta = (const float*)d_in[6];
  const float* W_out   = (const float*)d_in[7];
  (void)in_sizes; (void)n_in; (void)out_size; (void)ws_size;

  float*  h_all = (float*)d_out;                              // [(T+1), B, D]
  float*  outs  = h_all + (size_t)(T_STEPS + 1) * BD;         // [T, B, D]
  float2* stats = (float2*)d_ws;                              // [T*B, NGROUPS]

  dim3 grid(MROWS / 128, DIM / 128);
  dim3 block(256);

  // delta = sigmoid(x W_delta^T + b_delta) -> in place at h_all[1:]
  dsc_gemm_kernel<<<grid, block, 0, stream>>>(x, W_delta, b_delta,
                                              nullptr, nullptr,
                                              h_all + BD, 1);
  // cand = x W_x^T + b -> outs region (scratch until final GEMM)
  dsc_gemm_kernel<<<grid, block, 0, stream>>>(x, W_x, b,
                                              nullptr, nullptr,
                                              outs, 0);
  // sequential diagonal recurrence + group-softmax stats
  dsc_scan_kernel<<<(BATCH * NGROUPS * 32) / 256, block, 0, stream>>>(
      h0, r_h, h_all, outs, stats);
  // outs = compete(h) * silu(h W_out^T)
  dsc_gemm_kernel<<<grid, block, 0, stream>>>(h_all + BD, W_out, nullptr,
                                              h_all + BD, stats,
                                              outs, 2);
}